// GCNAE_46600395162290
// MI455X (gfx1250) — compile-verified
//
#include <hip/hip_runtime.h>

// ---------------------------------------------------------------------------
// GCN 3-layer forward for MI455X (gfx1250, wave32).
//   x:[N,15] f32, edge_index:[2,E] int32, W1/W2:[15,15], W3:[15,4], b1,b2:[15], b3:[4]
// Pipeline per layer: WMMA f32 16x16x4 GEMM -> self-loop init -> edge atomics.
// Node feature rows padded to 16 floats (64B): aligned float4 gathers and
// aligned v2f A-matrix loads in the GEMM.
// ---------------------------------------------------------------------------

typedef __attribute__((ext_vector_type(2))) float v2f;
typedef __attribute__((ext_vector_type(8))) float v8f;

#define THREADS 256

// ---------------- degree / normalization --------------------------------

__global__ void k_init_deg(float* __restrict__ deg, int n) {
    int i = blockIdx.x * blockDim.x + threadIdx.x;
    if (i < n) deg[i] = 1.0f;                 // self-loop contributes 1
}

__global__ void k_deg_edges(const int* __restrict__ dst, float* __restrict__ deg, int e) {
    int i = blockIdx.x * blockDim.x + threadIdx.x;
    if (i < e) unsafeAtomicAdd(&deg[dst[i]], 1.0f);
}

__global__ void k_finish_dis(float* __restrict__ dis, int n) {
    int i = blockIdx.x * blockDim.x + threadIdx.x;
    if (i < n) {
        float d = dis[i];
        dis[i] = (d > 0.0f) ? rsqrtf(d) : 0.0f;
    }
}

// ---------------- pad x [N,15] -> [N,16] (col 15 = 0) -------------------

__global__ void k_pad_x(const float* __restrict__ x, float* __restrict__ xp, int n) {
    int i = blockIdx.x * blockDim.x + threadIdx.x;   // over n*16
    if (i < n * 16) {
        int node = i >> 4, f = i & 15;
        xp[i] = (f < 15) ? x[(long)node * 15 + f] : 0.0f;
    }
}

// ---------------- WMMA GEMM: h[N,16] = f(in) @ Wpad ---------------------
// One wave computes a 16-node x 16-feature tile with 4 chained
// V_WMMA_F32_16X16X4_F32 (K=4 each, total K=16; col/row 15 zero-padded).
// Input rows are stride 16 with col 15 == 0.  If HAS_BIAS, the input
// transform is relu(in + bias) (previous layer's epilogue, folded in).

template <bool HAS_BIAS>
__global__ void k_gemm_wmma(const float* __restrict__ xin,
                            const float* __restrict__ bias_in,
                            const float* __restrict__ W, int wN,
                            float* __restrict__ hout, int n_nodes) {
    const int wave = blockIdx.x * (blockDim.x >> 5) + (threadIdx.x >> 5);
    const int lane = threadIdx.x & 31;
    const int row_base = wave * 16;
    if (row_base >= n_nodes) return;          // wave-uniform exit

    const int half = lane >> 4;               // 0: K offset {0,1}, 1: K offset {2,3}
    const int l15  = lane & 15;

    // B (W padded to 16x16): kblock kb, VGPR j holds K = kb*4 + j + 2*half, N = l15
    v2f bmat[4];
#pragma unroll
    for (int kb = 0; kb < 4; ++kb) {
        int k0 = kb * 4 + 2 * half;
        int k0c = (k0 < 15) ? k0 : 0;
        int k1c = (k0 + 1 < 15) ? (k0 + 1) : 0;
        int nc  = (l15 < wN) ? l15 : 0;
        float bx = W[k0c * wN + nc];
        float by = W[k1c * wN + nc];
        bmat[kb].x = (k0     < 15 && l15 < wN) ? bx : 0.0f;
        bmat[kb].y = (k0 + 1 < 15 && l15 < wN) ? by : 0.0f;
    }

    // A row for this lane: 8B-aligned pairs (k0 even)
    const v2f* xr2 = reinterpret_cast<const v2f*>(xin + (long)(row_base + l15) * 16);

    v8f c = {0.f, 0.f, 0.f, 0.f, 0.f, 0.f, 0.f, 0.f};
#pragma unroll
    for (int kb = 0; kb < 4; ++kb) {
        int k0 = kb * 4 + 2 * half;
        v2f raw = xr2[kb * 2 + half];          // elements {k0, k0+1}
        v2f a;
        if (HAS_BIAS) {
            int k1  = k0 + 1;
            int k1c = (k1 < 15) ? k1 : 0;
            float bx = bias_in[k0];            // k0 <= 14: in-bounds
            float by = bias_in[k1c];
            a.x = fmaxf(raw.x + bx, 0.0f);
            float ay = fmaxf(raw.y + by, 0.0f);
            a.y = (k1 < 15) ? ay : 0.0f;       // re-zero the K=15 pad
        } else {
            a = raw;                           // pad col already 0
        }
        c = __builtin_amdgcn_wmma_f32_16x16x4_f32(
                false, a, false, bmat[kb], (short)0, c, false, false);
    }

    // D: VGPR v -> row (v + 8*half), col l15 ; write all 16 cols (pad col = 0)
    float* hr = hout + (long)row_base * 16;
#pragma unroll
    for (int v = 0; v < 8; ++v) {
        hr[(v + 8 * half) * 16 + l15] = c[v];
    }
}

// ---------------- aggregation ------------------------------------------

// out[i][f] = h[i][f] * dis[i]^2   (self-loop term; also initializes buffer)
__global__ void k_agg_init(const float* __restrict__ h, const float* __restrict__ dis,
                           float* __restrict__ out, int n) {
    int i = blockIdx.x * blockDim.x + threadIdx.x;   // over n*16
    if (i < n * 16) {
        float s = dis[i >> 4];
        out[i] = h[i] * s * s;
    }
}

// out[dst] += h[src] * dis[src]*dis[dst]  for every edge (NF features)
template <int NF>
__global__ void k_agg_edges(const int* __restrict__ src, const int* __restrict__ dst,
                            const float* __restrict__ h, const float* __restrict__ dis,
                            float* __restrict__ out, int e) {
    int i = blockIdx.x * blockDim.x + threadIdx.x;
    if (i >= e) return;
    int s = src[i];
    int d = dst[i];
    float nrm = dis[s] * dis[d];
    const float4* hr = reinterpret_cast<const float4*>(h + (long)s * 16);
    float* orow = out + (long)d * 16;
    if (NF == 4) {
        float4 v0 = hr[0];
        unsafeAtomicAdd(&orow[0], v0.x * nrm);
        unsafeAtomicAdd(&orow[1], v0.y * nrm);
        unsafeAtomicAdd(&orow[2], v0.z * nrm);
        unsafeAtomicAdd(&orow[3], v0.w * nrm);
    } else {
        float4 v0 = hr[0], v1 = hr[1], v2 = hr[2], v3 = hr[3];
        unsafeAtomicAdd(&orow[0],  v0.x * nrm);
        unsafeAtomicAdd(&orow[1],  v0.y * nrm);
        unsafeAtomicAdd(&orow[2],  v0.z * nrm);
        unsafeAtomicAdd(&orow[3],  v0.w * nrm);
        unsafeAtomicAdd(&orow[4],  v1.x * nrm);
        unsafeAtomicAdd(&orow[5],  v1.y * nrm);
        unsafeAtomicAdd(&orow[6],  v1.z * nrm);
        unsafeAtomicAdd(&orow[7],  v1.w * nrm);
        unsafeAtomicAdd(&orow[8],  v2.x * nrm);
        unsafeAtomicAdd(&orow[9],  v2.y * nrm);
        unsafeAtomicAdd(&orow[10], v2.z * nrm);
        unsafeAtomicAdd(&orow[11], v2.w * nrm);
        unsafeAtomicAdd(&orow[12], v3.x * nrm);
        unsafeAtomicAdd(&orow[13], v3.y * nrm);
        unsafeAtomicAdd(&orow[14], v3.z * nrm);
        // col 15 stays zero -> skip
    }
}

// d_out[i*4+f] = agg[i*16+f] + b3[f]
__global__ void k_final(const float* __restrict__ agg, const float* __restrict__ b3,
                        float* __restrict__ out, int n) {
    int i = blockIdx.x * blockDim.x + threadIdx.x;   // over n*4
    if (i < n * 4) {
        out[i] = agg[(long)(i >> 2) * 16 + (i & 3)] + b3[i & 3];
    }
}

// ---------------------------------------------------------------------------

extern "C" void kernel_launch(void* const* d_in, const int* in_sizes, int n_in,
                              void* d_out, int out_size, void* d_ws, size_t ws_size,
                              hipStream_t stream) {
    const float* x   = (const float*)d_in[0];
    const int*   ei  = (const int*)d_in[1];     // [2, E]: first E = src, next E = dst
    const float* W1  = (const float*)d_in[3];
    const float* b1  = (const float*)d_in[4];
    const float* W2  = (const float*)d_in[5];
    const float* b2  = (const float*)d_in[6];
    const float* W3  = (const float*)d_in[7];
    const float* b3  = (const float*)d_in[8];
    float* out = (float*)d_out;

    const int n = in_sizes[0] / 15;             // 500000
    const int e = in_sizes[1] / 2;              // 16000000
    const int* src = ei;
    const int* dst = ei + e;

    // workspace: dis [n] | hA [n*16] | hB [n*16]
    char* ws = (char*)d_ws;
    float* dis = (float*)ws;
    float* hA  = (float*)(ws + (((size_t)n * 4 + 255) & ~(size_t)255));
    float* hB  = hA + (size_t)n * 16;

    const int gN   = (n + THREADS - 1) / THREADS;
    const int gE   = (e + THREADS - 1) / THREADS;
    const int g16N = (n * 16 + THREADS - 1) / THREADS;
    const int g4N  = (n * 4 + THREADS - 1) / THREADS;
    const int waves = (n + 15) / 16;
    const int gGemm = (waves + 7) / 8;          // 8 waves per 256-thread block

    // normalization: deg -> dis (in place)
    k_init_deg  <<<gN, THREADS, 0, stream>>>(dis, n);
    k_deg_edges <<<gE, THREADS, 0, stream>>>(dst, dis, e);
    k_finish_dis<<<gN, THREADS, 0, stream>>>(dis, n);

    // pad x into hB (stride 16, col 15 = 0); GEMM1 consumes hB before agg
    // overwrites it.
    k_pad_x<<<g16N, THREADS, 0, stream>>>(x, hB, n);

    // layer 1: h = xpad @ W1 ; agg -> hB
    k_gemm_wmma<false><<<gGemm, THREADS, 0, stream>>>(hB, nullptr, W1, 15, hA, n);
    k_agg_init        <<<g16N,  THREADS, 0, stream>>>(hA, dis, hB, n);
    k_agg_edges<15>   <<<gE,    THREADS, 0, stream>>>(src, dst, hA, dis, hB, e);

    // layer 2: input = relu(hB + b1)
    k_gemm_wmma<true> <<<gGemm, THREADS, 0, stream>>>(hB, b1, W2, 15, hA, n);
    k_agg_init        <<<g16N,  THREADS, 0, stream>>>(hA, dis, hB, n);
    k_agg_edges<15>   <<<gE,    THREADS, 0, stream>>>(src, dst, hA, dis, hB, e);

    // layer 3: input = relu(hB + b2), W3 is 15x4
    k_gemm_wmma<true> <<<gGemm, THREADS, 0, stream>>>(hB, b2, W3, 4, hA, n);
    k_agg_init        <<<g16N,  THREADS, 0, stream>>>(hA, dis, hB, n);
    k_agg_edges<4>    <<<gE,    THREADS, 0, stream>>>(src, dst, hA, dis, hB, e);

    // epilogue: + b3, pack [n,4]
    k_final<<<g4N, THREADS, 0, stream>>>(hB, b3, out, n);
}